// Attention_5549097746460
// MI455X (gfx1250) — compile-verified
//
#include <hip/hip_runtime.h>
#include <stdint.h>

typedef __attribute__((ext_vector_type(2))) float v2f;
typedef __attribute__((ext_vector_type(8))) float v8f;

#define EPS   1e-7f
#define NB    32      // batch
#define NT    4096    // steps
#define ND    512     // feature dim (D == E)
#define CHUNK_ROWS 128
#define WAVES 8
#define CHUNKS (NT / CHUNK_ROWS)   // 32

// ---------------------------------------------------------------------------
// Kernel A: yp[m,n] = sum_k y[m,k] * W[n,k]   (yp = y @ W^T), fp32 WMMA.
// One wave per 16x16 output tile; K consumed 4 at a time by
// V_WMMA_F32_16X16X4_F32.
//   A (16x4 f32) per-lane layout: lanes 0-15 hold M=lane, K={k,k+1};
//                                 lanes 16-31 hold M=lane-16, K={k+2,k+3}.
//   B (4x16 f32) mirrors with N in place of M; B[k,n] = W[n,k].
//   C/D (16x16 f32): VGPR r -> (M=r, N=lane) lanes 0-15, (M=r+8) lanes 16-31.
// ---------------------------------------------------------------------------
__global__ __launch_bounds__(32) void yp_gemm_wmma(const float* __restrict__ y,
                                                   const float* __restrict__ W,
                                                   float* __restrict__ yp) {
  const int lane = threadIdx.x;       // 0..31, no divergence -> EXEC all 1s
  const int half = lane >> 4;         // 0: K pair {k,k+1}, 1: {k+2,k+3}
  const int idx  = lane & 15;         // M (for A) / N (for B) within tile
  const int m0   = (blockIdx.x >> 5) * 16;   // 2 M-tiles (B=32)
  const int n0   = (blockIdx.x & 31) * 16;   // 32 N-tiles (D=512)

  const float* arow = y + (size_t)(m0 + idx) * ND;  // y[m, :]
  const float* brow = W + (size_t)(n0 + idx) * ND;  // W[n, :]

  v8f acc = {};
  for (int k0 = 0; k0 < ND; k0 += 4) {
    v2f a = *(const v2f*)(arow + k0 + 2 * half);
    v2f b = *(const v2f*)(brow + k0 + 2 * half);
    acc = __builtin_amdgcn_wmma_f32_16x16x4_f32(false, a, false, b,
                                                (short)0, acc, false, false);
  }

#pragma unroll
  for (int r = 0; r < 8; ++r) {
    const int m = m0 + r + 8 * half;
    yp[(size_t)m * ND + n0 + idx] = acc[r];
  }
}

// ---------------------------------------------------------------------------
// Kernel B (bandwidth-bound pass): stream x once with coalesced b128 loads.
// Block = 8 waves, owns (batch b, 128-row chunk of T). Each wave computes one
// full 512-float dot product per row: 32 lanes x float4 x 4 iterations, then
// a wave32 xor-shuffle reduction. Writes unnormalized a and a deterministic
// per-block partial sum.
// ---------------------------------------------------------------------------
__global__ __launch_bounds__(256) void score_kernel(const float* __restrict__ x,
                                                    const float* __restrict__ yp,
                                                    const float* __restrict__ bias,
                                                    const uint8_t* __restrict__ mask,
                                                    float* __restrict__ out,
                                                    float* __restrict__ partials) {
  const int b     = blockIdx.x / CHUNKS;
  const int chunk = blockIdx.x % CHUNKS;
  const int wave  = threadIdx.x >> 5;
  const int lane  = threadIdx.x & 31;

  // Per-lane fragment of yp[b,:]: lane covers d = i*128 + lane*4 .. +3
  float4 ypf[4];
#pragma unroll
  for (int i = 0; i < 4; ++i)
    ypf[i] = *(const float4*)(yp + (size_t)b * ND + i * 128 + lane * 4);

  const float bs = bias[0];
  float wave_sum = 0.0f;
  const int t0 = chunk * CHUNK_ROWS;

  for (int r = wave; r < CHUNK_ROWS; r += WAVES) {
    const int t = t0 + r;
    const float* xrow = x + ((size_t)b * NT + t) * ND;

    // hint next row for this wave into cache (global_prefetch_b8)
    if (r + WAVES < CHUNK_ROWS)
      __builtin_prefetch(xrow + (size_t)WAVES * ND + lane * 16, 0, 0);

    float acc = 0.0f;
#pragma unroll
    for (int i = 0; i < 4; ++i) {
      const float4 xv = *(const float4*)(xrow + i * 128 + lane * 4);
      acc = fmaf(xv.x, ypf[i].x, acc);
      acc = fmaf(xv.y, ypf[i].y, acc);
      acc = fmaf(xv.z, ypf[i].z, acc);
      acc = fmaf(xv.w, ypf[i].w, acc);
    }
    // wave32 reduction: after xor tree, every lane holds the row sum
#pragma unroll
    for (int off = 16; off > 0; off >>= 1)
      acc += __shfl_xor(acc, off, 32);

    const float e = acc + bs;
    const float a = mask[(size_t)b * NT + t] ? expf(tanhf(e)) : 0.0f;

    if (lane == 0) {
      out[(size_t)b * NT + t] = a;
      wave_sum += a;   // fixed iteration order -> deterministic
    }
  }

  __shared__ float s_ws[WAVES];
  if (lane == 0) s_ws[wave] = wave_sum;
  __syncthreads();
  if (threadIdx.x == 0) {
    float s = 0.0f;
#pragma unroll
    for (int w = 0; w < WAVES; ++w) s += s_ws[w];  // fixed order
    partials[(size_t)b * CHUNKS + chunk] = s;
  }
}

// ---------------------------------------------------------------------------
// Kernel C: per-batch deterministic reduction of 32 partials, then normalize.
// ---------------------------------------------------------------------------
__global__ __launch_bounds__(256) void normalize_kernel(float* __restrict__ out,
                                                        const float* __restrict__ partials) {
  const int b = blockIdx.x;
  __shared__ float s_inv;
  if (threadIdx.x == 0) {
    float s = 0.0f;
    for (int c = 0; c < CHUNKS; ++c) s += partials[(size_t)b * CHUNKS + c];
    s_inv = 1.0f / (s + EPS);
  }
  __syncthreads();
  const float inv = s_inv;
  for (int t = threadIdx.x; t < NT; t += 256)
    out[(size_t)b * NT + t] *= inv;
}

// ---------------------------------------------------------------------------
extern "C" void kernel_launch(void* const* d_in, const int* in_sizes, int n_in,
                              void* d_out, int out_size, void* d_ws, size_t ws_size,
                              hipStream_t stream) {
  (void)in_sizes; (void)n_in; (void)out_size; (void)ws_size;

  const float*   x    = (const float*)d_in[0];    // [32,4096,512]
  const float*   y    = (const float*)d_in[1];    // [32,512]
  const float*   W    = (const float*)d_in[2];    // [512,512]
  const float*   bias = (const float*)d_in[3];    // [1]
  const uint8_t* mask = (const uint8_t*)d_in[4];  // [32,4096] bool

  float* out      = (float*)d_out;                // [32,4096]
  float* yp       = (float*)d_ws;                 // 32*512 floats  (64 KB)
  float* partials = yp + NB * ND;                 // 32*32 floats   (4 KB)

  yp_gemm_wmma   <<<NB / 16 * (ND / 16), 32, 0, stream>>>(y, W, yp);
  score_kernel   <<<NB * CHUNKS, 256, 0, stream>>>(x, yp, bias, mask, out, partials);
  normalize_kernel<<<NB, 256, 0, stream>>>(out, partials);
}